// AuxiliaryGIN_84670985273386
// MI455X (gfx1250) — compile-verified
//
#include <hip/hip_runtime.h>
#include <hip/hip_bf16.h>

#define N_NODES 50000
#define N_EDGES 640000
#define D 128
#define LDS_STRIDE 132   // 128 + 4 pad: conflict-free strided fragment reads

typedef __attribute__((ext_vector_type(2))) float v2f;
typedef __attribute__((ext_vector_type(8))) float v8f;

// ---------------------------------------------------------------------------
// Zero a float buffer (float4 granularity).
// ---------------------------------------------------------------------------
__global__ void zero_f4(float4* __restrict__ p, int n4) {
  int i = blockIdx.x * blockDim.x + threadIdx.x;
  if (i < n4) {
    float4 z; z.x = 0.f; z.y = 0.f; z.z = 0.f; z.w = 0.f;
    p[i] = z;
  }
}

// ---------------------------------------------------------------------------
// Edge scatter-add: agg[dst[e]] += x[src[e]].
// 32 threads per edge, one float4 (16B) each -> coalesced row gather,
// 128 consecutive atomic floats per edge on the scatter side (L2 atomics).
// ---------------------------------------------------------------------------
__global__ __launch_bounds__(256) void scatter_add(
    const float* __restrict__ x, const int* __restrict__ src,
    const int* __restrict__ dst, float* __restrict__ agg, int nEdges) {
  int tid = blockIdx.x * blockDim.x + threadIdx.x;
  int e = tid >> 5;
  if (e >= nEdges) return;
  int c = (tid & 31) << 2;
  int s = __ldg(src + e);
  int d = __ldg(dst + e);
  const float4 v = *(const float4*)(x + (size_t)s * D + c);
  float* p = agg + (size_t)d * D + c;
  atomicAdd(p + 0, v.x);
  atomicAdd(p + 1, v.y);
  atomicAdd(p + 2, v.z);
  atomicAdd(p + 3, v.w);
}

// ---------------------------------------------------------------------------
// Fused GEMM: out = epilogue( (A [+ A2]) @ W + bias )
//   fuse==1: epilogue = relu( v * (gamma/sqrt(1+eps)) + beta )   (BN eval + ReLU)
//   fuse==0: epilogue = v                                        (bias only)
// A: [nRows,128] row-major, W: [128,128] row-major (W[k*128+n]).
// 4 waves/block, each wave computes a 16-row x 128-col stripe via
// V_WMMA_F32_16X16X4_F32 (8 accumulator tiles, 32 K-steps each = 256 wmma).
// In-place out==A (or out==A2) is safe: the full A tile is staged to LDS
// before any store, and blocks own disjoint row ranges.
// ---------------------------------------------------------------------------
__global__ __launch_bounds__(128) void gin_gemm(
    const float* __restrict__ A, const float* __restrict__ A2,
    const float* __restrict__ W, const float* __restrict__ bias,
    const float* __restrict__ gamma, const float* __restrict__ beta,
    float* __restrict__ out, int nRows, int fuse) {
  __shared__ float As[4][16 * LDS_STRIDE];

  const int wave = threadIdx.x >> 5;   // wave32
  const int lane = threadIdx.x & 31;
  const int row0 = (blockIdx.x * 4 + wave) * 16;

  float* as = As[wave];

  // Stage 16x128 A tile (with optional fused x+agg add); pad OOB rows with 0.
  for (int i = lane; i < 16 * 32; i += 32) {
    int r  = i >> 5;        // row within tile
    int c4 = (i & 31) << 2; // float4 column
    int row = row0 + r;
    float4 v; v.x = 0.f; v.y = 0.f; v.z = 0.f; v.w = 0.f;
    if (row < nRows) {
      v = *(const float4*)(A + (size_t)row * D + c4);
      if (A2 != nullptr) {
        float4 v2 = *(const float4*)(A2 + (size_t)row * D + c4);
        v.x += v2.x; v.y += v2.y; v.z += v2.z; v.w += v2.w;
      }
    }
    *(float4*)(as + r * LDS_STRIDE + c4) = v;
  }
  // Wave-local LDS producer/consumer: compiler inserts s_wait_dscnt.

  const int half = lane >> 4;   // 0: K pair {0,1}; 1: K pair {2,3}
  const int l16  = lane & 15;

  v8f acc[8];
#pragma unroll
  for (int nt = 0; nt < 8; ++nt) acc[nt] = (v8f)(0.0f);

  for (int k = 0; k < D; k += 4) {
    // A fragment (16x4 f32): lane half selects K offset pair.
    v2f a;
    a.x = as[l16 * LDS_STRIDE + k + 2 * half];
    a.y = as[l16 * LDS_STRIDE + k + 2 * half + 1];
#pragma unroll
    for (int nt = 0; nt < 8; ++nt) {
      int ncol = nt * 16 + l16;
      v2f b;  // B fragment (4x16 f32), column n = l16
      b.x = W[(k + 2 * half) * D + ncol];
      b.y = W[(k + 2 * half + 1) * D + ncol];
      acc[nt] = __builtin_amdgcn_wmma_f32_16x16x4_f32(
          /*neg_a=*/false, a, /*neg_b=*/false, b,
          /*c_mod=*/(short)0, acc[nt], /*reuse_a=*/false, /*reuse_b=*/false);
    }
  }

  const float inv = rsqrtf(1.0f + 1e-5f);
#pragma unroll
  for (int nt = 0; nt < 8; ++nt) {
    int col = nt * 16 + l16;
    float bb = bias[col];
    float g = 0.f, be = 0.f;
    if (fuse) { g = gamma[col] * inv; be = beta[col]; }
#pragma unroll
    for (int r = 0; r < 8; ++r) {
      int row = row0 + r + 8 * half;   // D layout: VGPR r -> M = r (+8 for hi half)
      if (row < nRows) {
        float v = acc[nt][r] + bb;
        if (fuse) v = fmaxf(v * g + be, 0.f);
        out[(size_t)row * D + col] = v;
      }
    }
  }
}

// ---------------------------------------------------------------------------
// Heads: per node, 128x(40+40+1+1) matvec + log_softmax / softmax / sigmoid.
// One 128-thread block per node.
// ---------------------------------------------------------------------------
__global__ __launch_bounds__(128) void heads_kernel(
    const float* __restrict__ h,
    const float* __restrict__ clsW, const float* __restrict__ clsB,
    const float* __restrict__ simW, const float* __restrict__ simB,
    const float* __restrict__ homW, const float* __restrict__ homB,
    const float* __restrict__ entW, const float* __restrict__ entB,
    float* __restrict__ out_main, float* __restrict__ out_sim,
    float* __restrict__ out_hom, float* __restrict__ out_ent) {
  __shared__ float hs[D];
  __shared__ float logits[80];
  __shared__ float red[4];

  const int node = blockIdx.x;
  const int t = threadIdx.x;
  hs[t] = h[(size_t)node * D + t];
  __syncthreads();

  if (t < 40) {
    float acc = clsB[t];
    for (int k = 0; k < D; ++k) acc += hs[k] * clsW[k * 40 + t];
    logits[t] = acc;
  } else if (t < 80) {
    int j = t - 40;
    float acc = simB[j];
    for (int k = 0; k < D; ++k) acc += hs[k] * simW[k * 40 + j];
    logits[t] = acc;
  } else if (t == 80) {
    float acc = homB[0];
    for (int k = 0; k < D; ++k) acc += hs[k] * homW[k];
    out_hom[node] = 1.0f / (1.0f + expf(-acc));
  } else if (t == 81) {
    float acc = entB[0];
    for (int k = 0; k < D; ++k) acc += hs[k] * entW[k];
    out_ent[node] = 1.0f / (1.0f + expf(-acc));
  }
  __syncthreads();

  if (t == 0) {  // log-softmax stats over cls logits
    float m = logits[0];
    for (int i = 1; i < 40; ++i) m = fmaxf(m, logits[i]);
    float s = 0.f;
    for (int i = 0; i < 40; ++i) s += expf(logits[i] - m);
    red[0] = m + logf(s);          // logsumexp
  } else if (t == 1) {  // softmax stats over sim logits
    float m = logits[40];
    for (int i = 41; i < 80; ++i) m = fmaxf(m, logits[i]);
    float s = 0.f;
    for (int i = 40; i < 80; ++i) s += expf(logits[i] - m);
    red[1] = m;
    red[2] = s;
  }
  __syncthreads();

  if (t < 40) {
    out_main[(size_t)node * 40 + t] = logits[t] - red[0];
  } else if (t < 80) {
    out_sim[(size_t)node * 40 + (t - 40)] = expf(logits[t] - red[1]) / red[2];
  }
}

// ---------------------------------------------------------------------------
extern "C" void kernel_launch(void* const* d_in, const int* in_sizes, int n_in,
                              void* d_out, int out_size, void* d_ws, size_t ws_size,
                              hipStream_t stream) {
  const float* x     = (const float*)d_in[0];
  const int* ei      = (const int*)d_in[1];
  const int* src     = ei;
  const int* dst     = ei + N_EDGES;
  const float* c1_W1 = (const float*)d_in[2];
  const float* c1_b1 = (const float*)d_in[3];
  const float* c1_g1 = (const float*)d_in[4];
  const float* c1_be1= (const float*)d_in[5];
  const float* c1_W2 = (const float*)d_in[6];
  const float* c1_b2 = (const float*)d_in[7];
  const float* c2_W1 = (const float*)d_in[8];
  const float* c2_b1 = (const float*)d_in[9];
  const float* c2_g1 = (const float*)d_in[10];
  const float* c2_be1= (const float*)d_in[11];
  const float* c2_W2 = (const float*)d_in[12];
  const float* c2_b2 = (const float*)d_in[13];
  const float* bn_g  = (const float*)d_in[14];
  const float* bn_b  = (const float*)d_in[15];
  const float* cls_W = (const float*)d_in[16];
  const float* cls_b = (const float*)d_in[17];
  const float* sim_W = (const float*)d_in[18];
  const float* sim_b = (const float*)d_in[19];
  const float* hom_W = (const float*)d_in[20];
  const float* hom_b = (const float*)d_in[21];
  const float* ent_W = (const float*)d_in[22];
  const float* ent_b = (const float*)d_in[23];

  float* out_main = (float*)d_out;                       // [50000,40]
  float* out_sim  = out_main + (size_t)N_NODES * 40;     // [50000,40]
  float* out_hom  = out_sim  + (size_t)N_NODES * 40;     // [50000]
  float* out_ent  = out_hom  + N_NODES;                  // [50000]

  const size_t NF = (size_t)N_NODES * D;                 // 6.4M floats
  float* agg  = (float*)d_ws;                            // buffer 0
  float* hbuf = agg + NF;                                // buffer 1

  const int n4 = (int)(NF / 4);
  dim3 zgrid((n4 + 255) / 256), zblk(256);
  dim3 sgrid((N_EDGES * 32 + 255) / 256), sblk(256);
  dim3 ggrid((N_NODES + 63) / 64), gblk(128);

  // ---- conv1 ----
  zero_f4<<<zgrid, zblk, 0, stream>>>((float4*)agg, n4);
  scatter_add<<<sgrid, sblk, 0, stream>>>(x, src, dst, agg, N_EDGES);
  // hbuf = relu(bn1((x + agg) @ W1 + b1))
  gin_gemm<<<ggrid, gblk, 0, stream>>>(x, agg, c1_W1, c1_b1, c1_g1, c1_be1,
                                       hbuf, N_NODES, 1);
  // hbuf = relu(bn_outer(hbuf @ W2 + b2))     (in-place)
  gin_gemm<<<ggrid, gblk, 0, stream>>>(hbuf, nullptr, c1_W2, c1_b2, bn_g, bn_b,
                                       hbuf, N_NODES, 1);

  // ---- conv2 ----
  zero_f4<<<zgrid, zblk, 0, stream>>>((float4*)agg, n4);
  scatter_add<<<sgrid, sblk, 0, stream>>>(hbuf, src, dst, agg, N_EDGES);
  // agg = relu(bn2((hbuf + agg) @ W1 + b1))   (in-place into agg)
  gin_gemm<<<ggrid, gblk, 0, stream>>>(hbuf, agg, c2_W1, c2_b1, c2_g1, c2_be1,
                                       agg, N_NODES, 1);
  // agg = agg @ W2 + b2                       (in-place, no activation)
  gin_gemm<<<ggrid, gblk, 0, stream>>>(agg, nullptr, c2_W2, c2_b2, nullptr, nullptr,
                                       agg, N_NODES, 0);

  // ---- heads ----
  heads_kernel<<<dim3(N_NODES), dim3(128), 0, stream>>>(
      agg, cls_W, cls_b, sim_W, sim_b, hom_W, hom_b, ent_W, ent_b,
      out_main, out_sim, out_hom, out_ent);
}